// TemporalSeparateAttention_25864293056657
// MI455X (gfx1250) — compile-verified
//
#include <hip/hip_runtime.h>

typedef __attribute__((ext_vector_type(2))) float v2f;
typedef __attribute__((ext_vector_type(8))) float v8f;

#define C_DIM 256
#define NHEAD 8
#define HD 32
#define B_DIM 2

__device__ __forceinline__ v8f wmma_f32(v2f a, v2f b, v8f c) {
  // D = A(16x4 f32) * B(4x16 f32) + C(16x16 f32)
  return __builtin_amdgcn_wmma_f32_16x16x4_f32(
      /*neg_a=*/false, a, /*neg_b=*/false, b,
      /*c_mod=*/(short)0, c, /*reuse_a=*/false, /*reuse_b=*/false);
}

// ---------------------------------------------------------------------------
// Elementwise add (q + q_pos, k + k_pos), float4 vectorized.
// ---------------------------------------------------------------------------
__global__ void add_kernel(const float* __restrict__ a, const float* __restrict__ b,
                           float* __restrict__ o, int n4) {
  int i = blockIdx.x * blockDim.x + threadIdx.x;
  if (i < n4) {
    float4 x = ((const float4*)a)[i];
    float4 y = ((const float4*)b)[i];
    ((float4*)o)[i] = make_float4(x.x + y.x, x.y + y.y, x.z + y.z, x.w + y.w);
  }
}

// ---------------------------------------------------------------------------
// Batched GEMM: Out[b, oRowOff+m, n] = sum_k A[b, aRowOff+m, k] * W[n, k]
// K = N = 256. One wave computes a 16x64 tile via 4 f32 WMMA accumulators.
// grid = (N/64, M/16, B), block = 32.
// ---------------------------------------------------------------------------
__global__ void __launch_bounds__(32)
gemm_bt(const float* __restrict__ A, int aRowOff, int aBatchStride,
        const float* __restrict__ W,
        float* __restrict__ Out, int oRowOff, int oBatchStride) {
  const int lane = threadIdx.x;
  const int half = lane >> 4;   // which half of the K-group this lane holds
  const int ln   = lane & 15;
  const int n0 = blockIdx.x * 64;
  const int m0 = blockIdx.y * 16;
  const int b  = blockIdx.z;

  const float* Ab  = A + (size_t)b * aBatchStride + (size_t)(aRowOff + m0 + ln) * C_DIM;
  const float* Wp0 = W + (size_t)(n0 + ln) * C_DIM;

  v8f acc[4] = {};
#pragma unroll 4
  for (int k = 0; k < C_DIM; k += 4) {
    const int kk = k + 2 * half;
    v2f a = *(const v2f*)(Ab + kk);          // A 16x4 chunk, A-layout
#pragma unroll
    for (int t = 0; t < 4; ++t) {
      v2f w = *(const v2f*)(Wp0 + (size_t)t * 16 * C_DIM + kk);  // W^T 4x16, B-layout
      acc[t] = wmma_f32(a, w, acc[t]);
    }
  }

  float* Ob = Out + (size_t)b * oBatchStride +
              (size_t)(oRowOff + m0 + 8 * half) * C_DIM + n0 + ln;
#pragma unroll
  for (int r = 0; r < 8; ++r)
#pragma unroll
    for (int t = 0; t < 4; ++t)
      Ob[(size_t)r * C_DIM + t * 16] = acc[t][r];
}

// ---------------------------------------------------------------------------
// Flash attention, one wave per (b, head, 16-query tile).
// Qh: [B, Qrows, 256], Kh/Vh: [B, Krows, 256] (head h uses cols h*32..h*32+31)
// mask: [1024, 3072]; row = qMaskRowOff + global q; col mapped through the
// sep's key gather: col = lk < kSplit ? lk+colOff0 : lk-kSplit+colOff1.
// O: [B, 1024, 256], rows written at oRowOff + q.
// ---------------------------------------------------------------------------
__global__ void __launch_bounds__(32)
flash_attn(const float* __restrict__ Qh, const float* __restrict__ Kh,
           const float* __restrict__ Vh, const float* __restrict__ mask,
           float* __restrict__ O,
           int Qrows, int Krows, int qMaskRowOff,
           int kSplit, int colOff0, int colOff1, int oRowOff) {
  __shared__ float pl[16 * 18];            // padded P tile (avoids bank conflicts)
  const int lane = threadIdx.x;
  const int half = lane >> 4;
  const int ln   = lane & 15;
  const int q0 = blockIdx.x * 16;
  const int h  = blockIdx.y;
  const int b  = blockIdx.z;
  const int n0 = h * HD;

  // Load Q tile [16 x 32] in A-layout: 8 chunks of 16x4.
  const float* Qb = Qh + ((size_t)b * Qrows + q0 + ln) * C_DIM + n0;
  v2f qa[8];
#pragma unroll
  for (int c = 0; c < 8; ++c) qa[c] = *(const v2f*)(Qb + 4 * c + 2 * half);

  v8f o0 = {}, o1 = {};                    // O accumulator [16 x 32], C-layout
  float m[8], l[8];
#pragma unroll
  for (int r = 0; r < 8; ++r) { m[r] = -3.0e38f; l[r] = 0.0f; }

  const float scale = 0.17677669529663687f;  // 1/sqrt(32)
  const float* Kb = Kh + (size_t)b * Krows * C_DIM + n0;
  const float* Vb = Vh + (size_t)b * Krows * C_DIM;
  const int maskRow0 = qMaskRowOff + q0;

  for (int kb = 0; kb < Krows; kb += 16) {
    // ---- S = Q * K^T (16x16), 8 WMMAs over hd=32 ----
    v8f s = {};
    const float* Kc = Kb + (size_t)(kb + ln) * C_DIM;
#pragma unroll
    for (int c = 0; c < 8; ++c) {
      v2f kv = *(const v2f*)(Kc + 4 * c + 2 * half);  // K^T 4x16 chunk, B-layout
      s = wmma_f32(qa[c], kv, s);
    }

    // additive mask column for this lane's key
    const int lk  = kb + ln;
    const int col = (lk < kSplit) ? (lk + colOff0) : (lk - kSplit + colOff1);

    // ---- online softmax; row r lives at (r + 8*half), cols across half-wave ----
    float p[8];
#pragma unroll
    for (int r = 0; r < 8; ++r) {
      float logit = s[r] * scale +
                    mask[(size_t)(maskRow0 + r + 8 * half) * 3072 + col];
      float mx = logit;
#pragma unroll
      for (int off = 8; off > 0; off >>= 1)
        mx = fmaxf(mx, __shfl_xor(mx, off, 16));
      float mnew = fmaxf(m[r], mx);
      float fac  = __expf(m[r] - mnew);
      float pe   = __expf(logit - mnew);
      float sum  = pe;
#pragma unroll
      for (int off = 8; off > 0; off >>= 1)
        sum += __shfl_xor(sum, off, 16);
      l[r] = l[r] * fac + sum;
      m[r] = mnew;
      o0[r] *= fac;
      o1[r] *= fac;
      p[r] = pe;
    }

    // ---- C-layout -> A-layout transpose of P through LDS ----
#pragma unroll
    for (int r = 0; r < 8; ++r)
      pl[(r + 8 * half) * 18 + ln] = p[r];
    __syncthreads();

    // ---- O += P * V, 4 k-subchunks x 2 N-tiles ----
#pragma unroll
    for (int c = 0; c < 4; ++c) {
      v2f pa = *(const v2f*)(&pl[ln * 18 + 4 * c + 2 * half]);  // P 16x4, A-layout
      const float* Vr = Vb + (size_t)(kb + 4 * c + 2 * half) * C_DIM + n0 + ln;
      v2f v0; v0.x = Vr[0];  v0.y = Vr[C_DIM];
      v2f v1; v1.x = Vr[16]; v1.y = Vr[C_DIM + 16];
      o0 = wmma_f32(pa, v0, o0);
      o1 = wmma_f32(pa, v1, o1);
    }
    __syncthreads();
  }

  // ---- normalize and store ----
  float* Ob = O + ((size_t)b * 1024 + oRowOff + q0 + 8 * half) * C_DIM + n0 + ln;
#pragma unroll
  for (int r = 0; r < 8; ++r) {
    float inv = 1.0f / l[r];
    Ob[(size_t)r * C_DIM]      = o0[r] * inv;
    Ob[(size_t)r * C_DIM + 16] = o1[r] * inv;
  }
}

// ---------------------------------------------------------------------------
extern "C" void kernel_launch(void* const* d_in, const int* in_sizes, int n_in,
                              void* d_out, int out_size, void* d_ws, size_t ws_size,
                              hipStream_t stream) {
  const float* query = (const float*)d_in[0];
  const float* key   = (const float*)d_in[1];
  const float* value = (const float*)d_in[2];
  const float* qpos  = (const float*)d_in[3];
  const float* kpos  = (const float*)d_in[4];
  const float* mask  = (const float*)d_in[5];
  const float* Wq0 = (const float*)d_in[6];
  const float* Wk0 = (const float*)d_in[7];
  const float* Wv0 = (const float*)d_in[8];
  const float* Wo0 = (const float*)d_in[9];
  const float* Wq1 = (const float*)d_in[10];
  const float* Wk1 = (const float*)d_in[11];
  const float* Wv1 = (const float*)d_in[12];
  const float* Wo1 = (const float*)d_in[13];
  float* out = (float*)d_out;

  // workspace carve-up (floats)
  float* ws = (float*)d_ws;
  size_t off = 0;
  float* qp  = ws + off; off += (size_t)B_DIM * 1024 * C_DIM;  // q + q_pos
  float* kp  = ws + off; off += (size_t)B_DIM * 3072 * C_DIM;  // k + k_pos
  float* Qh0 = ws + off; off += (size_t)B_DIM * 896  * C_DIM;
  float* Kh0 = ws + off; off += (size_t)B_DIM * 3072 * C_DIM;
  float* Vh0 = ws + off; off += (size_t)B_DIM * 3072 * C_DIM;
  float* Qh1 = ws + off; off += (size_t)B_DIM * 128  * C_DIM;
  float* Kh1 = ws + off; off += (size_t)B_DIM * 1920 * C_DIM;
  float* Vh1 = ws + off; off += (size_t)B_DIM * 1920 * C_DIM;
  float* Oat = qp;  // qp is dead after projections; reuse as attention output

  const dim3 blk(32);

  // stage 1: positional adds
  {
    int n4q = B_DIM * 1024 * C_DIM / 4;
    int n4k = B_DIM * 3072 * C_DIM / 4;
    add_kernel<<<(n4q + 255) / 256, 256, 0, stream>>>(query, qpos, qp, n4q);
    add_kernel<<<(n4k + 255) / 256, 256, 0, stream>>>(key,   kpos, kp, n4k);
  }

  // stage 2: projections (A @ W^T)
  gemm_bt<<<dim3(4,  896/16, B_DIM), blk, 0, stream>>>(qp,    0,    1024*C_DIM, Wq0, Qh0, 0,   896*C_DIM);
  gemm_bt<<<dim3(4, 3072/16, B_DIM), blk, 0, stream>>>(kp,    0,    3072*C_DIM, Wk0, Kh0, 0,   3072*C_DIM);
  gemm_bt<<<dim3(4, 3072/16, B_DIM), blk, 0, stream>>>(value, 0,    3072*C_DIM, Wv0, Vh0, 0,   3072*C_DIM);
  gemm_bt<<<dim3(4,  128/16, B_DIM), blk, 0, stream>>>(qp,    896,  1024*C_DIM, Wq1, Qh1, 0,   128*C_DIM);
  // sep1 key/value gather = [ego 2688:3072, agent 0:1536], two contiguous launches
  gemm_bt<<<dim3(4,  384/16, B_DIM), blk, 0, stream>>>(kp,    2688, 3072*C_DIM, Wk1, Kh1, 0,   1920*C_DIM);
  gemm_bt<<<dim3(4, 1536/16, B_DIM), blk, 0, stream>>>(kp,    0,    3072*C_DIM, Wk1, Kh1, 384, 1920*C_DIM);
  gemm_bt<<<dim3(4,  384/16, B_DIM), blk, 0, stream>>>(value, 2688, 3072*C_DIM, Wv1, Vh1, 0,   1920*C_DIM);
  gemm_bt<<<dim3(4, 1536/16, B_DIM), blk, 0, stream>>>(value, 0,    3072*C_DIM, Wv1, Vh1, 384, 1920*C_DIM);

  // stage 3: flash attention (wave per b,h,16-query tile)
  //   sep0: mask rows 0.., cols identity (split covers everything)
  flash_attn<<<dim3(896/16, NHEAD, B_DIM), blk, 0, stream>>>(
      Qh0, Kh0, Vh0, mask, Oat, 896, 3072, /*qMaskRowOff=*/0,
      /*kSplit=*/3072, /*colOff0=*/0, /*colOff1=*/0, /*oRowOff=*/0);
  //   sep1: mask rows 896.., cols = lk<384 ? 2688+lk : lk-384
  flash_attn<<<dim3(128/16, NHEAD, B_DIM), blk, 0, stream>>>(
      Qh1, Kh1, Vh1, mask, Oat, 128, 1920, /*qMaskRowOff=*/896,
      /*kSplit=*/384, /*colOff0=*/2688, /*colOff1=*/0, /*oRowOff=*/896);

  // stage 4: output projections into d_out
  gemm_bt<<<dim3(4,  896/16, B_DIM), blk, 0, stream>>>(Oat, 0,   1024*C_DIM, Wo0, out, 0,   1024*C_DIM);
  gemm_bt<<<dim3(4,  128/16, B_DIM), blk, 0, stream>>>(Oat, 896, 1024*C_DIM, Wo1, out, 896, 1024*C_DIM);
}